// GeneAttentionLayer_21792664060257
// MI455X (gfx1250) — compile-verified
//
#include <hip/hip_runtime.h>
#include <math.h>

// ---------------------------------------------------------------------------
// GeneAttentionLayer fused kernel for MI455X (gfx1250, wave32, WMMA + async DMA)
//   N=64 batches, C=23 chromosomes (padded to 32), G=20000 genes, H=64
// ---------------------------------------------------------------------------

#define N_BATCH 64
#define C_CHROM 23
#define C_PAD   32
#define G_GENES 20000
#define H_DIM   64
#define TILE_G  16
#define WAVES   8
#define THREADS 256
#define XPAD    4              // LDS row padding (floats) to avoid bank conflicts
#define XROW    (H_DIM + XPAD) // 68 floats = 272 B, multiple of 16 B

typedef __attribute__((ext_vector_type(2))) float v2f;
typedef __attribute__((ext_vector_type(8))) float v8f;

// WMMA wrapper: real builtin in the device pass; parse-only stub in the host
// pass (host never executes device code, it only needs the body to parse).
static __device__ inline v8f wmma_f32_16x16x4(v2f a, v2f b, v8f c) {
#if defined(__HIP_DEVICE_COMPILE__)
  return __builtin_amdgcn_wmma_f32_16x16x4_f32(
      /*neg_a=*/false, a, /*neg_b=*/false, b,
      /*c_mod=*/(short)0, c, /*reuse_a=*/false, /*reuse_b=*/false);
#else
  (void)a; (void)b;
  return c;
#endif
}

// Async global->LDS DMA, 16 bytes per lane (gfx1250 GLOBAL_LOAD_ASYNC_TO_LDS).
// lds_addr: raw LDS byte offset (low 32 bits of the generic pointer); tracked
// by ASYNCcnt, so consume with s_wait_asynccnt.
static __device__ inline void async_load_b128(unsigned lds_addr,
                                              const void* gaddr) {
#if defined(__HIP_DEVICE_COMPILE__)
  asm volatile("global_load_async_to_lds_b128 %0, %1, off"
               :: "v"(lds_addr), "v"(gaddr) : "memory");
#else
  (void)lds_addr; (void)gaddr;
#endif
}

// ---------------------------------------------------------------------------
// Kernel 0: one-hot (C,G) -> assign[g] (argmax over the 23 rows)
// ---------------------------------------------------------------------------
__global__ __launch_bounds__(256) void assign_kernel(
    const float* __restrict__ chrom, int* __restrict__ assign) {
  int g = blockIdx.x * 256 + threadIdx.x;
  if (g >= G_GENES) return;
  int   c    = 0;
  float best = -1.0f;
#pragma unroll 1
  for (int i = 0; i < C_CHROM; ++i) {
    float v = chrom[i * G_GENES + g];
    if (v > best) { best = v; c = i; }
  }
  assign[g] = c;
}

// ---------------------------------------------------------------------------
// Kernel 1: fused score -> softmax-like weight -> weighted accumulation
// One workgroup per batch n; 8 waves each own independent 16-gene tiles,
// double-buffered through LDS via async DMA.
// ---------------------------------------------------------------------------
__global__ __launch_bounds__(THREADS) void gene_attention_kernel(
    const float* __restrict__ x,          // (N, G, H)
    const float* __restrict__ attention,  // (C, H)
    const int*   __restrict__ assign,     // (G)
    float*       __restrict__ out)        // (N, C, H)
{
  __shared__ float lds_x[WAVES][2][TILE_G][XROW];   // double-buffered X tiles
  __shared__ float lds_s[WAVES][TILE_G][C_PAD + 1]; // per-wave score tile
  __shared__ float lds_w[WAVES][TILE_G];            // per-gene weight
  __shared__ int   lds_c[WAVES][TILE_G];            // per-gene chromosome
  __shared__ float lds_numer[C_PAD][H_DIM];         // block accumulator
  __shared__ float lds_denom[C_PAD];

  const int n     = blockIdx.x;
  const int tid   = threadIdx.x;
  const int wave  = tid >> 5;
  const int lane  = tid & 31;
  const int l15   = lane & 15;
  const int lhalf = (lane < 16) ? 0 : 1;  // which K-pair this lane holds

  // ---- init shared accumulators ----
  for (int i = tid; i < C_PAD * H_DIM; i += THREADS)
    (&lds_numer[0][0])[i] = 0.0f;
  if (tid < C_PAD) lds_denom[tid] = 0.0f;
  __syncthreads();

  // ---- hoist attention B-operands into registers (loop-invariant) ----
  // battn[ct][kk] = attn^T 4x16 chunk operand for this lane:
  //   column c = ct*16 + l15, rows h0 = kk*4 + lhalf*2 (+1); zero for c >= 23.
  v2f battn[2][16];
#pragma unroll
  for (int ct = 0; ct < 2; ++ct) {
    const int c = ct * 16 + l15;
#pragma unroll
    for (int kk = 0; kk < 16; ++kk) {
      const int h0 = kk * 4 + lhalf * 2;
      v2f b = (v2f){};
      if (c < C_CHROM) {
        b.x = attention[c * H_DIM + h0];
        b.y = attention[c * H_DIM + h0 + 1];
      }
      battn[ct][kk] = b;
    }
  }

  // ---- persistent per-wave accumulators: numer tiles (2 row x 4 col of 16x16)
  v8f dacc[2][4];
#pragma unroll
  for (int rt = 0; rt < 2; ++rt)
#pragma unroll
    for (int hc = 0; hc < 4; ++hc)
      dacc[rt][hc] = (v8f){};

  const int n_tiles = G_GENES / TILE_G;  // 1250, exact

  // per-lane LDS byte offsets for the async DMA destinations (both buffers)
  const int f     = lane;                           // reused per chunk below
  unsigned  ldsb[2];
#pragma unroll
  for (int b = 0; b < 2; ++b)
    ldsb[b] = (unsigned)(unsigned long long)(const void*)&lds_x[wave][b][0][0];
  (void)f;

  // issue the first tile for this wave
  int t0 = wave;
  if (t0 < n_tiles) {
    const char* src =
        (const char*)(x + ((size_t)n * G_GENES + t0 * TILE_G) * H_DIM);
#pragma unroll
    for (int i = 0; i < 8; ++i) {
      int fi = i * 32 + lane;  // float4 index inside the 16x64 tile
      async_load_b128(ldsb[0] + (unsigned)((fi >> 4) * XROW + (fi & 15) * 4) * 4u,
                      src + (size_t)fi * 16u);
    }
  }

  int buf = 0;
  for (int t = t0; t < n_tiles; t += WAVES) {
    const int g0 = t * TILE_G;

    // prefetch tile t+16 into GL2 so the next async DMA hits cache
    if (t + 2 * WAVES < n_tiles)
      __builtin_prefetch(
          x + ((size_t)n * G_GENES + (g0 + 2 * WAVES * TILE_G)) * H_DIM, 0, 1);

    // wait for tile t to land in lds_x[wave][buf]
    asm volatile("s_wait_asynccnt 0" ::: "memory");

    // kick off tile t+8 into the other buffer while we compute on this one
    if (t + WAVES < n_tiles) {
      const char* src = (const char*)(
          x + ((size_t)n * G_GENES + (g0 + WAVES * TILE_G)) * H_DIM);
#pragma unroll
      for (int i = 0; i < 8; ++i) {
        int fi = i * 32 + lane;
        async_load_b128(
            ldsb[buf ^ 1] + (unsigned)((fi >> 4) * XROW + (fi & 15) * 4) * 4u,
            src + (size_t)fi * 16u);
      }
    }

    if (lane < 16) lds_c[wave][lane] = assign[g0 + lane];
    asm volatile("s_wait_dscnt 0" ::: "memory");

    // ---- step 1: scores  S(16g x 16c) = Xtile(16x64) * A^T(64x16), 2 c-tiles
#pragma unroll
    for (int ct = 0; ct < 2; ++ct) {
      v8f sacc = (v8f){};
#pragma unroll
      for (int kk = 0; kk < 16; ++kk) {  // K=64 as 16 chunks of 4
        const int h0 = kk * 4 + lhalf * 2;
        v2f a;
        a.x = lds_x[wave][buf][l15][h0];      // A: gene-major 16x4
        a.y = lds_x[wave][buf][l15][h0 + 1];
        sacc = wmma_f32_16x16x4(a, battn[ct][kk], sacc);
      }
#pragma unroll
      for (int j = 0; j < 8; ++j)             // D: vgpr j -> row (lhalf*8+j)
        lds_s[wave][lhalf * 8 + j][ct * 16 + l15] = sacc[j];
    }
    asm volatile("s_wait_dscnt 0" ::: "memory");

    // ---- per-gene weight: w = exp(leaky_relu(score[g][c(g)]))
    if (lane < 16) {
      int   c  = lds_c[wave][lane];
      float s  = lds_s[wave][lane][c];
      float lr = (s > 0.0f) ? s : 0.2f * s;
      float wg = expf(lr);
      lds_w[wave][lane] = wg;
      atomicAdd(&lds_denom[c], wg);           // ds_add_f32
    }
    asm volatile("s_wait_dscnt 0" ::: "memory");

    // ---- step 2: numer(32c x 64h) += W(32c x 16g) * Xtile(16g x 64h)
#pragma unroll
    for (int kg = 0; kg < 4; ++kg) {          // 16 genes as 4 K-chunks of 4
      const int grow0 = kg * 4 + lhalf * 2;
      v2f bB[4];
#pragma unroll
      for (int hc = 0; hc < 4; ++hc) {        // B: (g x h) from X tile
        bB[hc].x = lds_x[wave][buf][grow0][hc * 16 + l15];
        bB[hc].y = lds_x[wave][buf][grow0 + 1][hc * 16 + l15];
      }
#pragma unroll
      for (int rt = 0; rt < 2; ++rt) {
        const int cm = rt * 16 + l15;         // this lane's chromosome row
        v2f aW;
        aW.x = (lds_c[wave][grow0]     == cm) ? lds_w[wave][grow0]     : 0.0f;
        aW.y = (lds_c[wave][grow0 + 1] == cm) ? lds_w[wave][grow0 + 1] : 0.0f;
#pragma unroll
        for (int hc = 0; hc < 4; ++hc)
          dacc[rt][hc] = wmma_f32_16x16x4(aW, bB[hc], dacc[rt][hc]);
      }
    }

    buf ^= 1;
  }

  // ---- reduce per-wave partial numerators into the block accumulator ----
#pragma unroll
  for (int rt = 0; rt < 2; ++rt)
#pragma unroll
    for (int hc = 0; hc < 4; ++hc)
#pragma unroll
      for (int j = 0; j < 8; ++j)
        atomicAdd(&lds_numer[rt * 16 + lhalf * 8 + j][hc * 16 + l15],
                  dacc[rt][hc][j]);

  __syncthreads();

  // ---- normalize and write out (N, C, H) ----
  for (int i = tid; i < C_CHROM * H_DIM; i += THREADS) {
    int   c = i >> 6, h = i & 63;
    float d = lds_denom[c];
    d = (d > 1e-10f) ? d : 1e-10f;
    out[((size_t)n * C_CHROM + c) * H_DIM + h] = lds_numer[c][h] / d;
  }
}

// ---------------------------------------------------------------------------
extern "C" void kernel_launch(void* const* d_in, const int* in_sizes, int n_in,
                              void* d_out, int out_size, void* d_ws,
                              size_t ws_size, hipStream_t stream) {
  const float* x         = (const float*)d_in[0];  // (N, G, H) f32
  const float* chrom     = (const float*)d_in[1];  // (C, G)    f32 one-hot
  const float* attention = (const float*)d_in[2];  // (C, H)    f32
  float*       out       = (float*)d_out;          // (N, C, H) f32
  int*         assign    = (int*)d_ws;             // G ints of scratch

  assign_kernel<<<(G_GENES + 255) / 256, 256, 0, stream>>>(chrom, assign);
  gene_attention_kernel<<<N_BATCH, THREADS, 0, stream>>>(x, attention, assign,
                                                         out);
}